// DenoisingPotential_89223650607245
// MI455X (gfx1250) — compile-verified
//
#include <hip/hip_runtime.h>
#include <hip/hip_bf16.h>

// ---------------------------------------------------------------------------
// DenoisingPotential on gfx1250:
//   P_k = A_k^T A_k  (precompute once, kernel 1, WMMA f32 16x16x4)
//   10 fused iterations of x += alpha * grad phi(x), register-resident per
//   16-point tile.  Pd = diff * P^T via WMMA f32 16x16x4; P_k staged into
//   LDS with double-buffered GLOBAL_LOAD_ASYNC_TO_LDS_B128 (ASYNCcnt).
// ---------------------------------------------------------------------------

typedef __attribute__((ext_vector_type(2))) float v2f;
typedef __attribute__((ext_vector_type(4))) float v4f;
typedef __attribute__((ext_vector_type(8))) float v8f;
typedef __attribute__((ext_vector_type(4))) int   v4i;

typedef __attribute__((address_space(1))) v4i* gv4i_p;   // global int4*
typedef __attribute__((address_space(3))) v4i* lv4i_p;   // LDS int4*

#define DP_K      32
#define DP_D      64
#define DP_NITER  10
#define DP_WAVES  8                    // waves per block
#define DP_PTS    (DP_WAVES * 16)      // points per block = 128
#define PSTRIDE   68                   // padded LDS row stride (bank-conflict free)
#define PBUF      (DP_D * PSTRIDE)     // 4352 floats per P buffer

// ---------------------------------------------------------------------------
// Kernel 1: P[k] = A[k]^T A[k].  One wave per 16x16 output tile.
// ---------------------------------------------------------------------------
__global__ __launch_bounds__(256) void dp_precision_kernel(
    const float* __restrict__ A, float* __restrict__ P)
{
    const int lane = threadIdx.x & 31;
    const int wid  = (blockIdx.x * blockDim.x + threadIdx.x) >> 5;  // 0..511
    const int k    = wid >> 4;
    const int t    = wid & 15;
    const int ti   = (t >> 2) * 16;          // output row-tile (i)
    const int tl   = (t & 3)  * 16;          // output col-tile (l)
    const int n    = lane & 15;
    const int off2 = (lane >> 4) * 2;        // K-offset within chunk (0 or 2)

    const float* Ak = A + k * DP_D * DP_D;

    v8f acc = {};
#pragma unroll
    for (int jc = 0; jc < 16; ++jc) {
        const int j0 = jc * 4 + off2;
        v2f a, b;
        a.x = Ak[(j0 + 0) * DP_D + ti + n];   // (A^T)[ti+n][j] = A[j][ti+n]
        a.y = Ak[(j0 + 1) * DP_D + ti + n];
        b.x = Ak[(j0 + 0) * DP_D + tl + n];   // A[j][tl+n]
        b.y = Ak[(j0 + 1) * DP_D + tl + n];
        acc = __builtin_amdgcn_wmma_f32_16x16x4_f32(
            false, a, false, b, (short)0, acc, false, false);
    }

    const int hi8 = (lane >> 4) * 8;
    float* Pk = P + k * DP_D * DP_D;
#pragma unroll
    for (int v = 0; v < 8; ++v)
        Pk[(ti + v + hi8) * DP_D + tl + n] = acc[v];
}

// ---------------------------------------------------------------------------
// Kernel 2: 10 fused gradient-ascent iterations, one wave per 16 points.
// ---------------------------------------------------------------------------
__global__ __launch_bounds__(256) void dp_langevin_kernel(
    const float* __restrict__ x_in,
    const float* __restrict__ c,
    const float* __restrict__ mu,
    const float* __restrict__ Pg,
    float* __restrict__ x_out,
    const float* __restrict__ alphap)
{
    // sBig is time-multiplexed:
    //   during the k-loop : two padded P_k buffers (double-buffered async DMA)
    //   at iter boundaries: per-wave x transpose scratch (buffers idle there)
    __shared__ __align__(16) float sBig[2 * PBUF];            // 34816 B
    __shared__ __align__(16) float sMu[DP_K * DP_D];          //  8192 B
    __shared__            float sC[DP_K];

    const int tid   = threadIdx.x;
    const int lane  = tid & 31;
    const int wave  = tid >> 5;
    const int n     = lane & 15;        // N index / point index (A-frag)
    const int hi    = lane >> 4;        // half-wave select
    const int off2  = hi * 2;           // K-chunk sub-offset (0 or 2)
    const int hi8   = hi * 8;
    const int tileBase = blockIdx.x * DP_PTS + wave * 16;

    const float alpha = *alphap;

    // Preload mu / c into LDS (visible after the first __syncthreads below).
    for (int i = tid; i < DP_K * DP_D; i += 256) sMu[i] = mu[i];
    if (tid < DP_K) sC[tid] = c[tid];

    // Load x tile in C-layout: xc[it][v] = x[tileBase + v + 8*hi][16*it + n]
    float xc[4][8];
#pragma unroll
    for (int it = 0; it < 4; ++it)
#pragma unroll
        for (int v = 0; v < 8; ++v)
            xc[it][v] = x_in[(tileBase + v + hi8) * DP_D + it * 16 + n];

    float* sXw = sBig + wave * 16 * PSTRIDE;   // per-wave transpose region
    v2f xa[16];

    // C-layout -> A-fragment layout transpose through (padded) LDS.
    auto transpose_to_xa = [&]() {
        __syncthreads();                       // drain all P-buffer readers
#pragma unroll
        for (int it = 0; it < 4; ++it)
#pragma unroll
            for (int v = 0; v < 8; ++v)
                sXw[(v + hi8) * PSTRIDE + it * 16 + n] = xc[it][v];
        __syncthreads();
#pragma unroll
        for (int jc = 0; jc < 16; ++jc) {
            const float* p = &sXw[n * PSTRIDE + jc * 4 + off2];
            xa[jc].x = p[0];
            xa[jc].y = p[1];
        }
    };

    // Async DMA of P_k into one LDS buffer: 256 threads x 4 x b128 = 16 KB.
    auto prefetch_Pk = [&](int k, float* dstBuf) {
        const float* src = Pg + k * DP_D * DP_D + (tid >> 2) * DP_D + (tid & 3) * 16;
        float*       dst = dstBuf + (tid >> 2) * PSTRIDE + (tid & 3) * 16;
#pragma unroll
        for (int t = 0; t < 4; ++t) {
            __builtin_amdgcn_global_load_async_to_lds_b128(
                (gv4i_p)(src + 4 * t),
                (lv4i_p)(dst + 4 * t),
                0, 0);
        }
    };

    transpose_to_xa();
    __syncthreads();                 // xa reads drained; sBig free for DMA

#pragma unroll 1
    for (int iter = 0; iter < DP_NITER; ++iter) {
        // Online-softmax state per owned point (b = v + 8*hi).
        float Mx[8], S[8], g[4][8];
#pragma unroll
        for (int v = 0; v < 8; ++v) { Mx[v] = -3.0e38f; S[v] = 0.0f; }
#pragma unroll
        for (int it = 0; it < 4; ++it)
#pragma unroll
            for (int v = 0; v < 8; ++v) g[it][v] = 0.0f;

        prefetch_Pk(0, sBig);        // fill buffer 0 for k = 0

#pragma unroll 1
        for (int k = 0; k < DP_K; ++k) {
            const int cur = k & 1;
            float* sP = sBig + cur * PBUF;

            __builtin_amdgcn_s_wait_asynccnt(0);   // own DMA into buf[cur] done
            __syncthreads();                       // whole tile visible to block

            if (k + 1 < DP_K)                      // overlap next DMA w/ compute
                prefetch_Pk(k + 1, sBig + (cur ^ 1) * PBUF);

            const float* muk = sMu + k * DP_D;

            // ---- Pd = diff * P^T via 64 WMMA f32 16x16x4 ----
            v8f acc[4];
#pragma unroll
            for (int it = 0; it < 4; ++it) acc[it] = (v8f){};
#pragma unroll
            for (int jc = 0; jc < 16; ++jc) {
                v2f d;
                d.x = xa[jc].x - muk[jc * 4 + off2 + 0];
                d.y = xa[jc].y - muk[jc * 4 + off2 + 1];
#pragma unroll
                for (int it = 0; it < 4; ++it) {
                    // B-frag: P^T[4jc+r][16it+n] = P[16it+n][4jc+r]
                    const float* pp = &sP[(it * 16 + n) * PSTRIDE + jc * 4 + off2];
                    v2f b;
                    b.x = pp[0];
                    b.y = pp[1];
                    acc[it] = __builtin_amdgcn_wmma_f32_16x16x4_f32(
                        false, d, false, b, (short)0, acc[it], false, false);
                }
            }

            // ---- quad_b = diff . Pd  (C-layout elementwise + half-wave reduce)
            float dmu[4];
#pragma unroll
            for (int it = 0; it < 4; ++it) dmu[it] = muk[it * 16 + n];

            float part[8];
#pragma unroll
            for (int v = 0; v < 8; ++v) {
                float p = 0.0f;
#pragma unroll
                for (int it = 0; it < 4; ++it)
                    p = fmaf(acc[it][v], xc[it][v] - dmu[it], p);
                part[v] = p;
            }
#pragma unroll
            for (int msk = 1; msk < 16; msk <<= 1)
#pragma unroll
                for (int v = 0; v < 8; ++v)
                    part[v] += __shfl_xor(part[v], msk, 32);

            // ---- online softmax accumulate: g += e * Pd, S += e ----
            const float ck = sC[k];
#pragma unroll
            for (int v = 0; v < 8; ++v) {
                const float s    = ck - 0.5f * part[v];
                const float newM = fmaxf(Mx[v], s);
                const float sc   = __expf(Mx[v] - newM);
                const float e    = __expf(s - newM);
                Mx[v] = newM;
                S[v]  = S[v] * sc + e;
#pragma unroll
                for (int it = 0; it < 4; ++it)
                    g[it][v] = g[it][v] * sc + e * acc[it][v];
            }
        } // k

        // ---- x <- x - alpha * g / S ----
#pragma unroll
        for (int v = 0; v < 8; ++v) {
            const float w = -alpha / S[v];
#pragma unroll
            for (int it = 0; it < 4; ++it)
                xc[it][v] = fmaf(w, g[it][v], xc[it][v]);
        }

        if (iter != DP_NITER - 1) {  // uniform branch: barrier-safe
            transpose_to_xa();       // P buffers idle here; aliasing is safe
            __syncthreads();         // xa reads drained before next DMA
        }
    } // iter

    // ---- write final x ----
#pragma unroll
    for (int it = 0; it < 4; ++it)
#pragma unroll
        for (int v = 0; v < 8; ++v)
            x_out[(tileBase + v + hi8) * DP_D + it * 16 + n] = xc[it][v];
}

// ---------------------------------------------------------------------------
extern "C" void kernel_launch(void* const* d_in, const int* in_sizes, int n_in,
                              void* d_out, int out_size, void* d_ws, size_t ws_size,
                              hipStream_t stream)
{
    const float* x     = (const float*)d_in[0];
    const float* c     = (const float*)d_in[1];
    const float* mu    = (const float*)d_in[2];
    const float* A     = (const float*)d_in[3];
    const float* alpha = (const float*)d_in[4];
    float*       out   = (float*)d_out;
    float*       P     = (float*)d_ws;          // K*D*D floats = 512 KB scratch

    const int B = in_sizes[0] / DP_D;

    // 512 waves = 32 k * 16 tiles -> 64 blocks of 8 waves.
    dp_precision_kernel<<<(DP_K * 16) / DP_WAVES, 256, 0, stream>>>(A, P);

    dp_langevin_kernel<<<B / DP_PTS, 256, 0, stream>>>(x, c, mu, P, out, alpha);
}